// NNOpenSetClassifier_63934883169002
// MI455X (gfx1250) — compile-verified
//
#include <hip/hip_runtime.h>

typedef float v2f __attribute__((ext_vector_type(2)));
typedef float v8f __attribute__((ext_vector_type(8)));

constexpr int   kB      = 8;
constexpr int   kN      = 16384;
constexpr int   kD      = 128;
constexpr int   kNCat   = 21;
constexpr float kThresh = 230.0f;

constexpr int kWavesPerWG = 4;
constexpr int kWGs        = 1024;
constexpr int kPixPerWave = kN / (kWavesPerWG * kWGs);   // 16384 / 4096 = 4

__global__ __launch_bounds__(kWavesPerWG * 32)
void nn_openset_wmma(const float* __restrict__ X,     // [8, N, 128]
                     const float* __restrict__ TM,    // [64, N, 128]
                     const int*   __restrict__ tcls,  // [64]
                     float*       __restrict__ out)
{
    const int lane = (int)(threadIdx.x & 31u);
    const int tl   = lane & 15;    // column (t) / row (m) selector
    const int hi   = lane >> 4;    // half selector (K split)
    const int wave = (int)blockIdx.x * kWavesPerWG + ((int)threadIdx.x >> 5);

    float* out_pred = out;                                     // [8][N][21]
    float* out_mask = out_pred + (size_t)kB * kN * kNCat;      // [8][N]
    float* out_ncm  = out_mask + (size_t)kB * kN;              // [8][N]
    float* out_mind = out_ncm  + (size_t)kB * kN;              // [8][N]
    float* out_nc   = out_mind + (size_t)kB * kN;              // [8][N]

    constexpr int kRS = kN * kD;                 // row stride (floats) between b rows / t rows

    // Per-lane 32-bit indices (max ~33.5M floats -> fits easily in int; byte offset < 2^31).
    // A rows 8..15 duplicate rows 0..7: their products land only in lanes 16..31 of the
    // accumulators (M=8..15), which the reduction never consumes.
    const int aoff = (tl & 7) * kRS + 2 * hi;
    const int toff = tl * kRS + 2 * hi;

    // Wave-uniform tile base pointers (selected as SADDR by the backend).
    const float* tb1 = TM + (size_t)16 * kRS;
    const float* tb2 = TM + (size_t)32 * kRS;
    const float* tb3 = TM + (size_t)48 * kRS;

    for (int p = 0; p < kPixPerWave; ++p) {
        const int n    = wave * kPixPerWave + p;
        const int base = n * kD;                 // uniform per wave

        const float* xp  = X   + base;
        const float* t0p = TM  + base;
        const float* t1p = tb1 + base;
        const float* t2p = tb2 + base;
        const float* t3p = tb3 + base;

        v8f acc0 = {0.f,0.f,0.f,0.f,0.f,0.f,0.f,0.f};
        v8f acc1 = {0.f,0.f,0.f,0.f,0.f,0.f,0.f,0.f};
        v8f acc2 = {0.f,0.f,0.f,0.f,0.f,0.f,0.f,0.f};
        v8f acc3 = {0.f,0.f,0.f,0.f,0.f,0.f,0.f,0.f};

        float x2p  = 0.f;
        float t2s0 = 0.f, t2s1 = 0.f, t2s2 = 0.f, t2s3 = 0.f;

        #pragma unroll 4
        for (int k0 = 0; k0 < kD; k0 += 4) {
            // lanes 0-15: K = k0,k0+1 ; lanes 16-31: K = k0+2,k0+3 (A and B share the split)
            const v2f a  = *(const v2f*)(xp  + aoff + k0);
            const v2f b0 = *(const v2f*)(t0p + toff + k0);
            const v2f b1 = *(const v2f*)(t1p + toff + k0);
            const v2f b2 = *(const v2f*)(t2p + toff + k0);
            const v2f b3 = *(const v2f*)(t3p + toff + k0);

            x2p  += a.x * a.x + a.y * a.y;
            t2s0 += b0.x * b0.x + b0.y * b0.y;
            t2s1 += b1.x * b1.x + b1.y * b1.y;
            t2s2 += b2.x * b2.x + b2.y * b2.y;
            t2s3 += b3.x * b3.x + b3.y * b3.y;

            acc0 = __builtin_amdgcn_wmma_f32_16x16x4_f32(false, a, false, b0, (short)0, acc0, false, false);
            acc1 = __builtin_amdgcn_wmma_f32_16x16x4_f32(false, a, false, b1, (short)0, acc1, false, false);
            acc2 = __builtin_amdgcn_wmma_f32_16x16x4_f32(false, a, false, b2, (short)0, acc2, false, false);
            acc3 = __builtin_amdgcn_wmma_f32_16x16x4_f32(false, a, false, b3, (short)0, acc3, false, false);
        }

        // Prefetch next pixel's rows (+512 B per template row): one byte per lane pulls the line.
        if (p + 1 < kPixPerWave) {
            __builtin_prefetch(xp + kD + aoff, 0, 1);
            #pragma unroll
            for (int off = 0; off < kD; off += 32) {
                __builtin_prefetch(t0p + kD + toff + off, 0, 1);
                __builtin_prefetch(t1p + kD + toff + off, 0, 1);
                __builtin_prefetch(t2p + kD + toff + off, 0, 1);
                __builtin_prefetch(t3p + kD + toff + off, 0, 1);
            }
        }

        // Each lane covered half the K range; its xor-16 partner covered the other half.
        const float x2full = x2p  + __shfl_xor(x2p,  16, 32);  // lane tl: x2[b=tl] (tl<8 valid)
        const float t2f0   = t2s0 + __shfl_xor(t2s0, 16, 32);  // lane: t2[t = tt*16 + tl]
        const float t2f1   = t2s1 + __shfl_xor(t2s1, 16, 32);
        const float t2f2   = t2s2 + __shfl_xor(t2s2, 16, 32);
        const float t2f3   = t2s3 + __shfl_xor(t2s3, 16, 32);

        // Per batch r (C/D layout: VGPR r, lanes 0-15 hold M=r): distances, min/argmin over t.
        #pragma unroll
        for (int r = 0; r < kB; ++r) {
            const float xb = __shfl(x2full, r, 32);   // broadcast x2[b=r] from lane r

            const float d0 = xb - 2.f * acc0[r] + t2f0;
            const float d1 = xb - 2.f * acc1[r] + t2f1;
            const float d2 = xb - 2.f * acc2[r] + t2f2;
            const float d3 = xb - 2.f * acc3[r] + t2f3;

            float md = d0; int mi = tl;
            if (d1 < md) { md = d1; mi = 16 + tl; }
            if (d2 < md) { md = d2; mi = 32 + tl; }
            if (d3 < md) { md = d3; mi = 48 + tl; }

            // butterfly min+argmin across the 16 data lanes (first-index tie-break)
            #pragma unroll
            for (int m = 1; m <= 8; m <<= 1) {
                const float od = __shfl_xor(md, m, 32);
                const int   oi = __shfl_xor(mi, m, 32);
                if (od < md || (od == md && oi < mi)) { md = od; mi = oi; }
            }
            md = __shfl(md, 0, 32);
            mi = __shfl(mi, 0, 32);

            const int  c   = tcls[mi];
            const bool msk = (md <= kThresh);
            const size_t obase = (size_t)r * kN + (size_t)n;

            if (lane < kNCat)
                out_pred[obase * kNCat + lane] = (msk && (lane == c)) ? 1.f : 0.f;
            else if (lane == 21) out_mask[obase] = msk ? 1.f : 0.f;
            else if (lane == 22) out_ncm[obase]  = (float)(msk ? c : (kNCat - 1));
            else if (lane == 23) out_mind[obase] = md;
            else if (lane == 24) out_nc[obase]   = (float)c;
        }
    }
}

extern "C" void kernel_launch(void* const* d_in, const int* in_sizes, int n_in,
                              void* d_out, int out_size, void* d_ws, size_t ws_size,
                              hipStream_t stream) {
    (void)in_sizes; (void)n_in; (void)out_size; (void)d_ws; (void)ws_size;
    const float* X    = (const float*)d_in[0];
    const float* TM   = (const float*)d_in[1];
    const int*   tcls = (const int*)d_in[2];
    float*       out  = (float*)d_out;

    dim3 grid(kWGs), block(kWavesPerWG * 32);
    hipLaunchKernelGGL(nn_openset_wmma, grid, block, 0, stream, X, TM, tcls, out);
}